// SPLFullLoss_10213432230325
// MI455X (gfx1250) — compile-verified
//
#include <hip/hip_runtime.h>
#include <cstdint>

// SPLFullLoss for B=32, N=4096, D=512 fp32.
//
// Roofline: 512 MiB streamed once -> ~23 us at 23.3 TB/s HBM; ~0.75 flop/byte,
// purely memory bound. One fused streaming pass produces per-row (over D) and
// per-column (over N) {sum x^2, sum r^2, sum x*r}; tiny finalize kernels apply
// the normalize formula and reduce deterministically (no atomics).
//
// CDNA5 paths (verified in round-1 histogram: wmma=8, async asm assembled):
//  - global_load_async_to_lds_b128, 4-deep double buffering, ASYNCcnt-throttled
//  - v_wmma_f32_16x16x4_f32 ones-matrix trick for deterministic block reductions
// Round-2 changes: raw ds_bpermute wave reductions (kills the shfl bounds-check
// VALU chains seen in round-1 asm), float2 packed FMA accumulators (V_PK_FMA_F32
// opportunity), async pipeline deepened 2 -> 4 buffers.

#define EPSF 1e-12f

constexpr int CB = 32;
constexpr int CN = 4096;
constexpr int CD = 512;
constexpr int ROWS_PER_TILE = 256;
constexpr int TILES_PER_B = CN / ROWS_PER_TILE;   // 16
constexpr int NBLK_A = CB * TILES_PER_B;          // 512
constexpr int NBLK_ROW = 64;

#if defined(__has_builtin)
#if __has_builtin(__builtin_amdgcn_s_wait_asynccnt)
#define WAIT_ASYNC(n) __builtin_amdgcn_s_wait_asynccnt(n)
#endif
#if __has_builtin(__builtin_amdgcn_wmma_f32_16x16x4_f32)
#define USE_WMMA_F32 1
#endif
#if __has_builtin(__builtin_amdgcn_ds_bpermute)
#define HAVE_BPERM 1
#endif
#endif
#ifndef WAIT_ASYNC
#define WAIT_ASYNC(n) asm volatile("s_wait_asynccnt %0" ::"i"(n) : "memory")
#endif

typedef __attribute__((ext_vector_type(2))) float v2f;
typedef __attribute__((ext_vector_type(8))) float v8f;

__device__ __forceinline__ void async_load16(unsigned ldsByteOff, const float* g) {
  // VGLOBAL async load: VDST = LDS byte address (per-lane), VADDR = 64-bit global addr.
  asm volatile("global_load_async_to_lds_b128 %0, %1, off"
               ::"v"(ldsByteOff), "v"((unsigned long long)(uintptr_t)g)
               : "memory");
}

// xor-butterfly wave32 sum with precomputed bpermute byte-indices (no bounds-check VALU).
__device__ __forceinline__ float waveSum32(float v, int a16, int a8, int a4, int a2, int a1) {
#ifdef HAVE_BPERM
  v += __uint_as_float((unsigned)__builtin_amdgcn_ds_bpermute(a16, (int)__float_as_uint(v)));
  v += __uint_as_float((unsigned)__builtin_amdgcn_ds_bpermute(a8, (int)__float_as_uint(v)));
  v += __uint_as_float((unsigned)__builtin_amdgcn_ds_bpermute(a4, (int)__float_as_uint(v)));
  v += __uint_as_float((unsigned)__builtin_amdgcn_ds_bpermute(a2, (int)__float_as_uint(v)));
  v += __uint_as_float((unsigned)__builtin_amdgcn_ds_bpermute(a1, (int)__float_as_uint(v)));
#else
  for (int off = 16; off; off >>= 1) v += __shfl_xor(v, off, 32);
#endif
  return v;
}

// Deterministic reduction of 256 per-thread partials. red[] must hold >= 256 floats.
__device__ __forceinline__ float blockReduce256(float v, float* red) {
  const int tid = threadIdx.x;
  red[tid] = v;
  __syncthreads();
#ifdef USE_WMMA_F32
  if (tid < 32) {
    const int lane = tid;
    v2f ones;
    ones.x = 1.0f;
    ones.y = 1.0f;
    v8f acc = {};
    // A = ones(16x4) => D[m][n] = sum_k B[k][n] + C[m][n]; all 64 B-slots
    // (2 VGPRs x 32 lanes) hold distinct partials, so 4 chunks reduce 256 values.
#pragma unroll
    for (int c = 0; c < 4; ++c) {
      v2f bv;
      bv.x = red[c * 64 + lane * 2 + 0];
      bv.y = red[c * 64 + lane * 2 + 1];
      acc = __builtin_amdgcn_wmma_f32_16x16x4_f32(
          false, ones, false, bv, (short)0, acc, false, false);
    }
    // Rows of D are identical; summing acc[0] over 32 lanes gives 2x the total.
    float s = acc[0];
#pragma unroll
    for (int off = 16; off; off >>= 1) s += __shfl_xor(s, off, 32);
    if (lane == 0) red[0] = 0.5f * s;
  }
  __syncthreads();
#else
  for (int stride = 128; stride; stride >>= 1) {
    if (tid < stride) red[tid] += red[tid + stride];
    __syncthreads();
  }
#endif
  const float r = red[0];
  __syncthreads();
  return r;
}

// ---------------------------------------------------------------------------
// Kernel A: fused streaming pass. Grid = 512 blocks (one per (b, 256-row tile)),
// 256 threads = 8 waves; each wave owns 32 rows, each lane 16 fixed column
// positions. Data is staged through LDS with async loads, 4-deep pipelined.
// ---------------------------------------------------------------------------
__global__ __launch_bounds__(256) void splMainPass(const float* __restrict__ x,
                                                   const float* __restrict__ ref,
                                                   float* __restrict__ rowStats,
                                                   float* __restrict__ colPart) {
  __shared__ float smem[12288];  // 48 KB: staging (first 32 KB) then col-combine (all 48 KB)
  const int tid = threadIdx.x;
  const int wave = tid >> 5;
  const int lane = tid & 31;
  const int tile = blockIdx.x;      // 0..511
  const int b = tile >> 4;
  const int rowBase = (tile & 15) * ROWS_PER_TILE + wave * 32;
  const size_t gbase = ((size_t)b * CN + rowBase) * CD + (size_t)(lane * 4);

  // 4 KB per wave: 4 bufs x (128 x-floats + 128 r-floats)
  float* stg = &smem[wave * 1024];
  const unsigned ldsBase = (unsigned)(uintptr_t)stg + (unsigned)(lane * 16);

  // precomputed bpermute byte-indices for the xor butterfly
  const int a16 = ((lane ^ 16) << 2), a8 = ((lane ^ 8) << 2), a4 = ((lane ^ 4) << 2),
            a2 = ((lane ^ 2) << 2), a1 = ((lane ^ 1) << 2);

  v2f cxx[8], crr[8], cxr[8];
#pragma unroll
  for (int i = 0; i < 8; ++i) {
    cxx[i] = (v2f)(0.f);
    crr[i] = (v2f)(0.f);
    cxr[i] = (v2f)(0.f);
  }

  auto issue = [&](int s) {
    const int row = s >> 2, it = s & 3;
    const size_t off = gbase + (size_t)row * CD + (size_t)(it * 128);
    const unsigned l = ldsBase + (unsigned)((s & 3) * 1024);
    async_load16(l, x + off);
    async_load16(l + 512u, ref + off);
  };

  v2f rx2 = (v2f)(0.f), rr2 = (v2f)(0.f), rm2 = (v2f)(0.f);
  issue(0);
  issue(1);
  issue(2);
  for (int s = 0; s < 128; ++s) {          // 32 rows x 4 segments
    if (s + 3 < 128) {
      issue(s + 3);
      WAIT_ASYNC(6);                       // 3 younger pairs may stay outstanding
    } else {
      WAIT_ASYNC(0);                       // 3-step tail: drain
    }
    const int buf = s & 3, it = s & 3;
    const v2f* px = (const v2f*)&stg[buf * 256 + lane * 4];
    const v2f* pr = (const v2f*)&stg[buf * 256 + 128 + lane * 4];
    const v2f x0 = px[0], x1 = px[1];
    const v2f r0 = pr[0], r1 = pr[1];
    const int q = it * 2;
    cxx[q + 0] = x0 * x0 + cxx[q + 0];     // packed FMA candidates (V_PK_FMA_F32)
    cxx[q + 1] = x1 * x1 + cxx[q + 1];
    crr[q + 0] = r0 * r0 + crr[q + 0];
    crr[q + 1] = r1 * r1 + crr[q + 1];
    cxr[q + 0] = x0 * r0 + cxr[q + 0];
    cxr[q + 1] = x1 * r1 + cxr[q + 1];
    rx2 = x0 * x0 + rx2;
    rx2 = x1 * x1 + rx2;
    rr2 = r0 * r0 + rr2;
    rr2 = r1 * r1 + rr2;
    rm2 = x0 * r0 + rm2;
    rm2 = x1 * r1 + rm2;
    if (it == 3) {                          // row finished: cross-lane reduce
      float rxx = rx2.x + rx2.y;
      float rrr = rr2.x + rr2.y;
      float rxr = rm2.x + rm2.y;
      rxx = waveSum32(rxx, a16, a8, a4, a2, a1);
      rrr = waveSum32(rrr, a16, a8, a4, a2, a1);
      rxr = waveSum32(rxr, a16, a8, a4, a2, a1);
      if (lane == 0) {
        const size_t ro = ((size_t)b * CN + rowBase + (s >> 2)) * 3;
        rowStats[ro + 0] = rxx;
        rowStats[ro + 1] = rrr;
        rowStats[ro + 2] = rxr;
      }
      rx2 = (v2f)(0.f);
      rr2 = (v2f)(0.f);
      rm2 = (v2f)(0.f);
    }
  }

  __syncthreads();  // staging dead everywhere; reuse smem for column combine
#pragma unroll
  for (int it = 0; it < 4; ++it) {
#pragma unroll
    for (int h = 0; h < 2; ++h) {
      const int d = it * 128 + lane * 4 + h * 2;
      smem[(wave * 512 + d + 0) * 3 + 0] = cxx[it * 2 + h].x;
      smem[(wave * 512 + d + 1) * 3 + 0] = cxx[it * 2 + h].y;
      smem[(wave * 512 + d + 0) * 3 + 1] = crr[it * 2 + h].x;
      smem[(wave * 512 + d + 1) * 3 + 1] = crr[it * 2 + h].y;
      smem[(wave * 512 + d + 0) * 3 + 2] = cxr[it * 2 + h].x;
      smem[(wave * 512 + d + 1) * 3 + 2] = cxr[it * 2 + h].y;
    }
  }
  __syncthreads();
  for (int d = tid; d < CD; d += 256) {     // combine 8 waves in fixed order
    float sxx = 0.f, srr = 0.f, sxr = 0.f;
#pragma unroll
    for (int w = 0; w < 8; ++w) {
      const int o = (w * 512 + d) * 3;
      sxx += smem[o + 0];
      srr += smem[o + 1];
      sxr += smem[o + 2];
    }
    const size_t co = ((size_t)tile * CD + d) * 3;
    colPart[co + 0] = sxx;
    colPart[co + 1] = srr;
    colPart[co + 2] = sxr;
  }
}

// Kernel B: per batch, fold the 16 tile partials per column, apply normalize
// formula, reduce 512 column contributions -> batchColSum[b].
__global__ __launch_bounds__(256) void splColFinal(const float* __restrict__ colPart,
                                                   float* __restrict__ batchColSum) {
  __shared__ float red[256];
  const int b = blockIdx.x;
  const int tid = threadIdx.x;
  float partial = 0.f;
  for (int d = tid; d < CD; d += 256) {
    float sxx = 0.f, srr = 0.f, sxr = 0.f;
#pragma unroll
    for (int t = 0; t < TILES_PER_B; ++t) {
      const size_t o = (((size_t)(b * TILES_PER_B + t)) * CD + d) * 3;
      sxx += colPart[o + 0];
      srr += colPart[o + 1];
      sxr += colPart[o + 2];
    }
    const float dx = fmaxf(sqrtf(sxx), EPSF);
    const float dr = fmaxf(sqrtf(srr), EPSF);
    partial += sxr / (dx * dr);
  }
  const float tot = blockReduce256(partial, red);
  if (tid == 0) batchColSum[b] = tot;
}

// Kernel C: apply normalize formula to all 131072 rows, reduce per block.
__global__ __launch_bounds__(256) void splRowFinal(const float* __restrict__ rowStats,
                                                   float* __restrict__ blockRowSum) {
  __shared__ float red[256];
  const int tid = threadIdx.x;
  float partial = 0.f;
  for (int r = blockIdx.x * 256 + tid; r < CB * CN; r += NBLK_ROW * 256) {
    const size_t o = (size_t)r * 3;
    const float xx = rowStats[o + 0];
    const float rr = rowStats[o + 1];
    const float xr = rowStats[o + 2];
    const float dx = fmaxf(sqrtf(xx), EPSF);
    const float dr = fmaxf(sqrtf(rr), EPSF);
    partial += xr / (dx * dr);
  }
  const float tot = blockReduce256(partial, red);
  if (tid == 0) blockRowSum[blockIdx.x] = tot;
}

// Kernel D: fold tiny arrays in fixed order, apply scaling, write scalar.
__global__ void splFinal(const float* __restrict__ blockRowSum,
                         const float* __restrict__ batchColSum,
                         float* __restrict__ out) {
  float rt = 0.f, ct = 0.f;
  for (int i = 0; i < NBLK_ROW; ++i) rt += blockRowSum[i];
  for (int i = 0; i < CB; ++i) ct += batchColSum[i];
  out[0] = -(rt / (float)CN + ct / (float)CD) / (float)CB;
}

extern "C" void kernel_launch(void* const* d_in, const int* in_sizes, int n_in,
                              void* d_out, int out_size, void* d_ws, size_t ws_size,
                              hipStream_t stream) {
  (void)in_sizes; (void)n_in; (void)out_size; (void)ws_size;
  const float* x = (const float*)d_in[0];
  const float* ref = (const float*)d_in[1];
  float* ws = (float*)d_ws;
  // ws layout (floats): rowStats[3*B*N] | colPart[3*512*512] | batchColSum[32] | blockRowSum[64]
  float* rowStats = ws;
  float* colPart = rowStats + (size_t)3 * CB * CN;
  float* batchColSum = colPart + (size_t)3 * NBLK_A * CD;
  float* blockRowSum = batchColSum + CB;

  splMainPass<<<NBLK_A, 256, 0, stream>>>(x, ref, rowStats, colPart);
  splColFinal<<<CB, 256, 0, stream>>>(colPart, batchColSum);
  splRowFinal<<<NBLK_ROW, 256, 0, stream>>>(rowStats, blockRowSum);
  splFinal<<<1, 1, 0, stream>>>(blockRowSum, batchColSum, (float*)d_out);
}